// AttentionHead_43284680409109
// MI455X (gfx1250) — compile-verified
//
#include <hip/hip_runtime.h>

typedef __attribute__((ext_vector_type(2))) float v2f;
typedef __attribute__((ext_vector_type(8))) float v8f;

#define BATCH 4
#define SEQ   2048
#define DIN   1024
#define DOUT  64
#define ROWS  (BATCH * SEQ)   // 8192

static __device__ __forceinline__ v8f wmma_f32(v2f a, v2f b, v8f c) {
    // D = A(16x4) * B(4x16) + C(16x16), all fp32
    return __builtin_amdgcn_wmma_f32_16x16x4_f32(
        /*neg_a=*/false, a, /*neg_b=*/false, b,
        /*c_mod=*/(short)0, c, /*reuse_a=*/false, /*reuse_b=*/false);
}

// Async DMA: copies 16 bytes/lane from the global address sbase + voff to the LDS byte offset lds_byte_off, ASYNCcnt-tracked.
static __device__ __forceinline__ void async_ld_b128(unsigned lds_byte_off, unsigned voff,
                                                     const void* sbase) {
    asm volatile("global_load_async_to_lds_b128 %0, %1, %2 offset:0"
                 :: "v"(lds_byte_off), "v"(voff), "s"(sbase) : "memory");
}
static __device__ __forceinline__ void wait_async0() {
    asm volatile("s_wait_asynccnt 0" ::: "memory");
}

// ---------------------------------------------------------------------------
// Kernel 1: Q/K/V projections. Block = 128 threads (4 waves), one 16-row tile
// of X per block; wave w owns N-tile w for all three matrices. X k-chunks are
// double-buffered in LDS via async DMA, overlapped with the WMMA k-loop.
// LDS: 2 buffers x 16 rows x 68 floats (pad) = 8704 B (dynamic, base 0).
// ---------------------------------------------------------------------------
__global__ __launch_bounds__(128) void qkv_proj_kernel(
    const float* __restrict__ X,
    const float* __restrict__ Wq, const float* __restrict__ bq,
    const float* __restrict__ Wk, const float* __restrict__ bk,
    const float* __restrict__ Wv, const float* __restrict__ bv,
    float* __restrict__ Q, float* __restrict__ K, float* __restrict__ V)
{
    extern __shared__ float smem[];               // [2][16*68]

    const int tid   = threadIdx.x;
    const int lane  = tid & 31;
    const int wv    = tid >> 5;            // 0..3 -> N-tile
    const int rowb  = blockIdx.x * 16;
    const int mm    = lane & 15;
    const int koff  = (lane >> 4) << 1;    // 0 or 2
    const int halfo = (lane >> 4) * 8;
    const int n     = wv * 16 + mm;

    // per-thread async coords: 2 x b128 per 16x64 chunk (256 vec4 / 128 thr)
    auto issue_chunk = [&](int kc, int buf) {
        #pragma unroll
        for (int i = 0; i < 2; ++i) {
            const int idx = tid * 2 + i;          // 0..255
            const int r   = idx >> 4;
            const int c   = (idx & 15) * 4;
            const unsigned lds  = (unsigned)((buf * 1088 + r * 68 + c) * 4);
            const unsigned voff = (unsigned)((((rowb + r) * DIN) + kc * 64 + c) * 4);
            async_ld_b128(lds, voff, X);
        }
    };

    issue_chunk(0, 0);                            // prologue

    v8f aq = {}, ak = {}, av = {};

    for (int kc = 0; kc < DIN / 64; ++kc) {
        wait_async0();                            // my chunk-kc loads landed
        __syncthreads();                          // everyone's landed; old buffer consumed
        if (kc + 1 < DIN / 64) issue_chunk(kc + 1, (kc + 1) & 1);

        const float* xs = smem + (kc & 1) * 1088;
        const int k0 = kc * 64;

        #pragma unroll
        for (int ks = 0; ks < 16; ++ks) {
            v2f a;
            a.x = xs[mm * 68 + ks * 4 + koff];
            a.y = xs[mm * 68 + ks * 4 + koff + 1];
            const int krow = k0 + ks * 4 + koff;
            v2f b;
            b.x = Wq[(size_t)krow * DOUT + n];
            b.y = Wq[(size_t)(krow + 1) * DOUT + n];
            aq = wmma_f32(a, b, aq);
            b.x = Wk[(size_t)krow * DOUT + n];
            b.y = Wk[(size_t)(krow + 1) * DOUT + n];
            ak = wmma_f32(a, b, ak);
            b.x = Wv[(size_t)krow * DOUT + n];
            b.y = Wv[(size_t)(krow + 1) * DOUT + n];
            av = wmma_f32(a, b, av);
        }
    }

    const float bqv = bq[n], bkv = bk[n], bvv = bv[n];
    #pragma unroll
    for (int r = 0; r < 8; ++r) {
        const size_t row = (size_t)rowb + halfo + r;   // C-layout: M = r + 8*(lane>=16)
        Q[row * DOUT + n] = aq[r] + bqv;
        K[row * DOUT + n] = ak[r] + bkv;
        V[row * DOUT + n] = av[r] + bvv;
    }
}

// ---------------------------------------------------------------------------
// Kernel 2: flash-style attention. Block = 64 threads (2 waves); each wave
// owns a 16-query tile; the block shares double-buffered async-DMA'd K/V
// 16x64 tiles in LDS. Online softmax with wave32 shuffle reductions.
// LDS (dynamic, base 0), floats:
//   buf b (b=0,1): K at b*2176, V at b*2176+1088 (16x68 each)
//   ps: per-wave 16x17 P scratch at 4352 + wv*272   -> total 4896 f = 19584 B
// ---------------------------------------------------------------------------
__global__ __launch_bounds__(64) void attn_kernel(
    const float* __restrict__ Q, const float* __restrict__ K,
    const float* __restrict__ V,
    const unsigned char* __restrict__ mask,   // bool [B,S,S], true = masked out
    float* __restrict__ out)
{
    extern __shared__ float smem[];

    const int tid   = threadIdx.x;
    const int lane  = tid & 31;
    const int wv    = tid >> 5;               // 0..1 -> q-tile within block
    const int b     = blockIdx.x;
    const int qbase = blockIdx.y * 32 + wv * 16;
    const int mm    = lane & 15;
    const int koff  = (lane >> 4) << 1;
    const int halfo = (lane >> 4) * 8;
    const float scale = 0.022097086912079608f;   // 1/sqrt(2048) (repo quirk: sqrt(S))

    const float* Kb = K + (size_t)b * SEQ * DOUT;
    const float* Vb = V + (size_t)b * SEQ * DOUT;

    // per-thread async coords: 4 x b128 per matrix per 16x64 tile
    auto issue_tile = [&](int kt, int buf) {
        #pragma unroll
        for (int i = 0; i < 4; ++i) {
            const int idx = tid + i * 64;          // 0..255
            const int r   = idx >> 4;
            const int c   = (idx & 15) * 4;
            const unsigned ldsK = (unsigned)((buf * 2176 + r * 68 + c) * 4);
            const unsigned voff = (unsigned)(((kt * 16 + r) * DOUT + c) * 4);
            async_ld_b128(ldsK,              voff, Kb);
            async_ld_b128(ldsK + 1088u * 4u, voff, Vb);
        }
    };

    // Q fragments in A-layout, scale folded in
    v2f qa[16];
    const size_t qrowg = (size_t)b * SEQ + qbase + mm;
    #pragma unroll
    for (int ks = 0; ks < 16; ++ks) {
        qa[ks].x = Q[qrowg * DOUT + ks * 4 + koff] * scale;
        qa[ks].y = Q[qrowg * DOUT + ks * 4 + koff + 1] * scale;
    }

    float mrun[8], lrun[8];
    #pragma unroll
    for (int r = 0; r < 8; ++r) { mrun[r] = -3.0e38f; lrun[r] = 0.0f; }
    v8f acc[4];
    {
        v8f z = {};
        #pragma unroll
        for (int nt = 0; nt < 4; ++nt) acc[nt] = z;
    }

    issue_tile(0, 0);                              // prologue

    for (int kt = 0; kt < SEQ / 16; ++kt) {
        wait_async0();                             // my tile-kt DMA landed
        __syncthreads();                           // everyone's landed; old buffer consumed
        if (kt + 1 < SEQ / 16) issue_tile(kt + 1, (kt + 1) & 1);

        const float* kts = smem + (kt & 1) * 2176;
        const float* vts = kts + 1088;
        float*       ps  = smem + 4352 + wv * 272;
        const int kbase  = kt * 16;

        // S-tile = Q * K^T (scaled)
        v8f s = {};
        #pragma unroll
        for (int ks = 0; ks < 16; ++ks) {
            v2f bf;
            bf.x = kts[mm * 68 + ks * 4 + koff];
            bf.y = kts[mm * 68 + ks * 4 + koff + 1];
            s = wmma_f32(qa[ks], bf, s);
        }

        // mask + online softmax statistics (row = r + halfo, col = mm)
        float cf[8];
        const int ncol = kbase + mm;
        #pragma unroll
        for (int r = 0; r < 8; ++r) {
            const int qrow = qbase + halfo + r;
            const unsigned char mk = mask[((size_t)b * SEQ + qrow) * SEQ + ncol];
            const float sv = mk ? -1.0e9f : s[r];
            float mx = sv;
            mx = fmaxf(mx, __shfl_xor(mx, 1, 32));
            mx = fmaxf(mx, __shfl_xor(mx, 2, 32));
            mx = fmaxf(mx, __shfl_xor(mx, 4, 32));
            mx = fmaxf(mx, __shfl_xor(mx, 8, 32));
            const float mnew = fmaxf(mrun[r], mx);
            const float corr = __expf(mrun[r] - mnew);
            const float p    = __expf(sv - mnew);
            float rs = p;
            rs += __shfl_xor(rs, 1, 32);
            rs += __shfl_xor(rs, 2, 32);
            rs += __shfl_xor(rs, 4, 32);
            rs += __shfl_xor(rs, 8, 32);
            lrun[r] = lrun[r] * corr + rs;
            mrun[r] = mnew;
            cf[r]   = corr;
            ps[(halfo + r) * 17 + mm] = p;         // C-layout -> LDS [M][N]
        }
        __syncthreads();   // uniform across waves; makes P visible for transposed re-read

        // rescale accumulators by row-wise correction
        #pragma unroll
        for (int nt = 0; nt < 4; ++nt) {
            #pragma unroll
            for (int r = 0; r < 8; ++r) acc[nt][r] *= cf[r];
        }

        // context += P(16x16) * V_tile(16x64)
        #pragma unroll
        for (int ks = 0; ks < 4; ++ks) {
            v2f pa;
            pa.x = ps[mm * 17 + ks * 4 + koff];
            pa.y = ps[mm * 17 + ks * 4 + koff + 1];
            #pragma unroll
            for (int nt = 0; nt < 4; ++nt) {
                v2f vb;
                vb.x = vts[(ks * 4 + koff) * 68 + nt * 16 + mm];
                vb.y = vts[(ks * 4 + koff + 1) * 68 + nt * 16 + mm];
                acc[nt] = wmma_f32(pa, vb, acc[nt]);
            }
        }
    }

    // normalize and store
    #pragma unroll
    for (int r = 0; r < 8; ++r) {
        const float inv = 1.0f / lrun[r];
        const size_t orow = (size_t)b * SEQ + qbase + halfo + r;
        #pragma unroll
        for (int nt = 0; nt < 4; ++nt)
            out[orow * DOUT + nt * 16 + mm] = acc[nt][r] * inv;
    }
}

// ---------------------------------------------------------------------------
extern "C" void kernel_launch(void* const* d_in, const int* in_sizes, int n_in,
                              void* d_out, int out_size, void* d_ws, size_t ws_size,
                              hipStream_t stream) {
    const float* X            = (const float*)d_in[0];
    const unsigned char* mask = (const unsigned char*)d_in[1];  // jax bool -> 1 byte/elem
    const float* Wq = (const float*)d_in[2];
    const float* bq = (const float*)d_in[3];
    const float* Wk = (const float*)d_in[4];
    const float* bk = (const float*)d_in[5];
    const float* Wv = (const float*)d_in[6];
    const float* bv = (const float*)d_in[7];
    float* out = (float*)d_out;

    float* Qb = (float*)d_ws;                      // 2 MB each, 6 MB total
    float* Kb = Qb + (size_t)ROWS * DOUT;
    float* Vb = Kb + (size_t)ROWS * DOUT;

    qkv_proj_kernel<<<ROWS / 16, 128, 2 * 1088 * sizeof(float), stream>>>(
        X, Wq, bq, Wk, bk, Wv, bv, Qb, Kb, Vb);
    attn_kernel<<<dim3(BATCH, SEQ / 32), 64, 4896 * sizeof(float), stream>>>(
        Qb, Kb, Vb, mask, out);

    (void)in_sizes; (void)n_in; (void)out_size; (void)ws_size;
}